// GNNEncoder_66709432041587
// MI455X (gfx1250) — compile-verified
//
#include <hip/hip_runtime.h>

typedef float v2f __attribute__((ext_vector_type(2)));
typedef float v8f __attribute__((ext_vector_type(8)));

// ---------------------------------------------------------------------------
// Utility: grid-stride float4 zero
// ---------------------------------------------------------------------------
__global__ void zero_kernel(float4* __restrict__ p, long n4) {
    long i = (long)blockIdx.x * blockDim.x + threadIdx.x;
    long stride = (long)gridDim.x * blockDim.x;
    float4 z = make_float4(0.f, 0.f, 0.f, 0.f);
    for (; i < n4; i += stride) p[i] = z;
}

// ---------------------------------------------------------------------------
// Edge counts per (relation, dst) and inverse mean denominators
// ---------------------------------------------------------------------------
__global__ void count_kernel(const int* __restrict__ dst,
                             const int* __restrict__ etype,
                             int* __restrict__ cnt, int E, int N) {
    int e = blockIdx.x * blockDim.x + threadIdx.x;
    if (e >= E) return;
    atomicAdd(&cnt[etype[e] * N + dst[e]], 1);
}

__global__ void inv_kernel(const int* __restrict__ cnt,
                           float* __restrict__ inv, int n) {
    int i = blockIdx.x * blockDim.x + threadIdx.x;
    if (i >= n) return;
    int c = cnt[i];
    inv[i] = 1.0f / (float)(c > 0 ? c : 1);
}

// ---------------------------------------------------------------------------
// Embedding gather: x[i, :] = emb[node_indices[i], :]   (float4 granularity)
// ---------------------------------------------------------------------------
__global__ void gather_kernel(const float* __restrict__ emb,
                              const int* __restrict__ idx,
                              float* __restrict__ x, int N, int d4) {
    int j = blockIdx.x * blockDim.x + threadIdx.x;
    if (j >= N * d4) return;
    int i = j / d4, q = j - i * d4;
    ((float4*)x)[(size_t)i * d4 + q] =
        ((const float4*)emb)[(size_t)idx[i] * d4 + q];
}

// ---------------------------------------------------------------------------
// Scatter-add: agg[etype[e]][dst[e]] += x[src[e]]   (float4 reads, f32 atomics)
// ---------------------------------------------------------------------------
template<int D4>
__global__ void scatter_kernel(const float* __restrict__ xin,
                               const int* __restrict__ src,
                               const int* __restrict__ dst,
                               const int* __restrict__ etype,
                               float* __restrict__ agg, int E, int N) {
    int j = blockIdx.x * blockDim.x + threadIdx.x;
    if (j >= E * D4) return;
    int e = j / D4, q = j - e * D4;
    int s = src[e], dn = dst[e], r = etype[e];
    float4 v = ((const float4*)xin)[(size_t)s * D4 + q];
    float* base = agg + ((size_t)r * N + dn) * (D4 * 4) + q * 4;
    atomicAdd(base + 0, v.x);
    atomicAdd(base + 1, v.y);
    atomicAdd(base + 2, v.z);
    atomicAdd(base + 3, v.w);
}

// ---------------------------------------------------------------------------
// WMMA fp32 GEMM over 9 K-segments (self/root + 8 relations).
//   C[M,NC] = A0 @ B0 + sum_r (inv_r * Agg_r) @ W_r  (+bias, opt relu, opt mask)
// Block: 256 threads (8 waves). Tile: BM=128, BN=128, BK=16.
// Wave grid 2x4; each wave owns a 64x32 patch = 4x2 tiles of 16x16 f32 acc.
// V_WMMA_F32_16X16X4_F32 (fp32 in/accumulate) for bitwise-safe math.
// Double-buffered LDS (one barrier per K-step, global loads a step ahead);
// padded LDS strides: A rows stride 20 (conflict-free 16-row column reads,
// b64-aligned), B rows stride 136 (lane halves hit disjoint bank groups).
// ---------------------------------------------------------------------------
template<bool RELU, bool MASK>
__global__ __launch_bounds__(256)
void gemm_kernel(const float* __restrict__ A0,    // [M, K0]
                 const float* __restrict__ Agg,   // [8, M, K0]
                 const float* __restrict__ B0,    // [K0, NC] root
                 const float* __restrict__ W,     // [8, K0, NC]
                 const float* __restrict__ bias,  // [NC]
                 const float* __restrict__ inv,   // [8*M]
                 const float* __restrict__ mask,  // [M] (MASK only)
                 float* __restrict__ C,           // [M, NC]
                 int M, int K0, int NC) {
    constexpr int BM = 128, BN = 128, BK = 16;
    constexpr int LDA = BK + 4;    // 20 floats: bank-conflict-free, 8B-aligned
    constexpr int LDB = BN + 8;    // 136 floats: lane halves -> disjoint banks
    __shared__ float lsA[2][BM * LDA];
    __shared__ float lsB[2][BK * LDB];

    const int tid  = threadIdx.x;
    const int lane = tid & 31;
    const int wid  = tid >> 5;
    const int wm   = (wid >> 2) * 64;   // wave M offset: 0 / 64
    const int wn   = (wid & 3) * 32;    // wave N offset: 0 / 32 / 64 / 96
    const int rowBase = blockIdx.y * BM;
    const int colBase = blockIdx.x * BN;

    const int lm = lane & 15;           // m (A) / n (B) within 16x16 tile
    const int kh = (lane >> 4) * 2;     // K sub-offset per lane half: 0 / 2
    const int rh = (lane >> 4) * 8;     // C/D row offset per lane half

    // Per-thread staging assignments (256 threads move 2+2 float4 per step)
    const int ar0 = tid >> 2,        ac0 = (tid & 3) * 4;          // A row/col
    const int ar1 = (tid + 256) >> 2;                              // second A row
    const int br0 = tid >> 5,        bc0 = (tid & 31) * 4;         // B row/col
    const int br1 = ((tid + 256) >> 5);                            // second B row

    const v8f vz = {0.f, 0.f, 0.f, 0.f, 0.f, 0.f, 0.f, 0.f};
    v8f acc[4][2];
#pragma unroll
    for (int tm = 0; tm < 4; ++tm)
#pragma unroll
        for (int tn = 0; tn < 2; ++tn) acc[tm][tn] = vz;

    const int stepsPerSeg = K0 / BK;
    const int totalSteps  = 9 * stepsPerSeg;

    float4 pa0, pa1, pb0, pb1;   // prefetch registers (global -> LDS pipeline)

    // Issue global loads for step s into prefetch registers (scale applied)
    auto load_step = [&](int s) {
        int seg = s / stepsPerSeg;
        int k0  = (s - seg * stepsPerSeg) * BK;
        const float* Aseg = (seg == 0) ? A0 : Agg + (size_t)(seg - 1) * M * K0;
        const float* Bseg = (seg == 0) ? B0 : W + (size_t)(seg - 1) * K0 * NC;
        pa0 = *(const float4*)(Aseg + (size_t)(rowBase + ar0) * K0 + k0 + ac0);
        pa1 = *(const float4*)(Aseg + (size_t)(rowBase + ar1) * K0 + k0 + ac0);
        pb0 = *(const float4*)(Bseg + (size_t)(k0 + br0) * NC + colBase + bc0);
        pb1 = *(const float4*)(Bseg + (size_t)(k0 + br1) * NC + colBase + bc0);
        if (seg != 0) {
            const float* invSeg = inv + (size_t)(seg - 1) * M;
            float s0 = invSeg[rowBase + ar0];
            float s1 = invSeg[rowBase + ar1];
            pa0.x *= s0; pa0.y *= s0; pa0.z *= s0; pa0.w *= s0;
            pa1.x *= s1; pa1.y *= s1; pa1.z *= s1; pa1.w *= s1;
        }
    };
    auto store_step = [&](int buf) {
        *(float4*)&lsA[buf][ar0 * LDA + ac0] = pa0;
        *(float4*)&lsA[buf][ar1 * LDA + ac0] = pa1;
        *(float4*)&lsB[buf][br0 * LDB + bc0] = pb0;
        *(float4*)&lsB[buf][br1 * LDB + bc0] = pb1;
    };

    // Pipeline prologue
    load_step(0);
    store_step(0);

    for (int t = 0; t < totalSteps; ++t) {
        const int cur = t & 1;
        if (t + 1 < totalSteps) load_step(t + 1);   // globals in flight
        __syncthreads();                            // LDS buffer cur fully written

#pragma unroll
        for (int kk = 0; kk < BK; kk += 4) {
            v2f afr[4];
#pragma unroll
            for (int tm = 0; tm < 4; ++tm)
                afr[tm] = *(const v2f*)&lsA[cur][(wm + tm * 16 + lm) * LDA + kk + kh];
            v2f bfr[2];
#pragma unroll
            for (int tn = 0; tn < 2; ++tn) {
                int col = wn + tn * 16 + lm;
                v2f b;
                b.x = lsB[cur][(kk + kh) * LDB + col];
                b.y = lsB[cur][(kk + kh + 1) * LDB + col];
                bfr[tn] = b;
            }
#pragma unroll
            for (int tm = 0; tm < 4; ++tm)
#pragma unroll
                for (int tn = 0; tn < 2; ++tn)
                    acc[tm][tn] = __builtin_amdgcn_wmma_f32_16x16x4_f32(
                        false, afr[tm], false, bfr[tn],
                        (short)0, acc[tm][tn], false, false);
        }

        if (t + 1 < totalSteps) store_step(cur ^ 1);   // safe: other buffer
    }

    // ---- epilogue: bias (+relu) (+mask), coalesced 16-lane row segments ----
#pragma unroll
    for (int tm = 0; tm < 4; ++tm) {
#pragma unroll
        for (int tn = 0; tn < 2; ++tn) {
            int col = colBase + wn + tn * 16 + lm;
            float b = bias[col];
#pragma unroll
            for (int v = 0; v < 8; ++v) {
                int row = rowBase + wm + tm * 16 + v + rh;
                float val = acc[tm][tn][v] + b;
                if (RELU) val = fmaxf(val, 0.0f);
                if (MASK) val *= mask[row];
                C[(size_t)row * NC + col] = val;
            }
        }
    }
}

// ---------------------------------------------------------------------------
// Host-side orchestration
// ---------------------------------------------------------------------------
static inline int cdiv(long a, int b) { return (int)((a + b - 1) / b); }

extern "C" void kernel_launch(void* const* d_in, const int* in_sizes, int n_in,
                              void* d_out, int out_size, void* d_ws, size_t ws_size,
                              hipStream_t stream) {
    const int* node_idx = (const int*)d_in[0];
    const int* edge_index = (const int*)d_in[1];
    const int* edge_type = (const int*)d_in[2];
    const float* mask = (const float*)d_in[4];
    const float* emb = (const float*)d_in[5];
    const float* W1 = (const float*)d_in[6];
    const float* root1 = (const float*)d_in[7];
    const float* b1 = (const float*)d_in[8];
    const float* W2 = (const float*)d_in[9];
    const float* root2 = (const float*)d_in[10];
    const float* b2 = (const float*)d_in[11];

    const int N = in_sizes[0];             // 16384
    const int E = in_sizes[2];             // 262144
    const int hid = in_sizes[8];           // 768
    const int emb_dim = in_sizes[7] / hid; // 256
    const int out_dim = in_sizes[11];      // 768
    const int R = 8;

    const int* src = edge_index;
    const int* dst = edge_index + E;

    // Workspace layout
    char* ws = (char*)d_ws;
    float* x   = (float*)ws;                         ws += (size_t)N * emb_dim * 4;
    float* h   = (float*)ws;                         ws += (size_t)N * hid * 4;
    int*   cnt = (int*)ws;                           ws += (size_t)R * N * 4;
    float* inv = (float*)ws;                         ws += (size_t)R * N * 4;
    float* agg = (float*)ws;  // R*N*hid floats, reused by both layers

    const int T = 256;

    // 1) relation counts + inverse mean denominators
    zero_kernel<<<1024, T, 0, stream>>>((float4*)cnt, (long)R * N * 4 / 16);
    count_kernel<<<cdiv(E, T), T, 0, stream>>>(dst, edge_type, cnt, E, N);
    inv_kernel<<<cdiv((long)R * N, T), T, 0, stream>>>(cnt, inv, R * N);

    // 2) embedding gather
    gather_kernel<<<cdiv((long)N * (emb_dim / 4), T), T, 0, stream>>>(
        emb, node_idx, x, N, emb_dim / 4);

    // 3) layer-1 scatter aggregation (d = 256)
    zero_kernel<<<4096, T, 0, stream>>>((float4*)agg, (long)R * N * emb_dim / 4);
    scatter_kernel<64><<<cdiv((long)E * 64, T), T, 0, stream>>>(
        x, src, dst, edge_type, agg, E, N);

    // 4) layer-1 WMMA GEMM: h = relu([x | mean-agg] @ [root1; W1] + b1)
    {
        dim3 grid(hid / 128, N / 128);
        gemm_kernel<true, false><<<grid, T, 0, stream>>>(
            x, agg, root1, W1, b1, inv, nullptr, h, N, emb_dim, hid);
    }

    // 5) layer-2 scatter aggregation (d = 768)
    zero_kernel<<<8192, T, 0, stream>>>((float4*)agg, (long)R * N * hid / 4);
    scatter_kernel<192><<<cdiv((long)E * 192, T), T, 0, stream>>>(
        h, src, dst, edge_type, agg, E, N);

    // 6) layer-2 WMMA GEMM: out = ([h | mean-agg] @ [root2; W2] + b2) * mask
    {
        dim3 grid(out_dim / 128, N / 128);
        gemm_kernel<false, true><<<grid, T, 0, stream>>>(
            h, agg, root2, W2, b2, inv, mask, (float*)d_out, N, hid, out_dim);
    }
}